// MOEDispatcher_51616916963600
// MI455X (gfx1250) — compile-verified
//
#include <hip/hip_runtime.h>
#include <stdint.h>

typedef float v2f __attribute__((ext_vector_type(2)));
typedef float v4f __attribute__((ext_vector_type(4)));
typedef float v8f __attribute__((ext_vector_type(8)));

#if defined(__AMDGCN__)
#if __has_builtin(__builtin_amdgcn_wmma_f32_16x16x4_f32)
#define USE_WMMA 1
#endif
#endif
#ifndef USE_WMMA
#define USE_WMMA 0
#endif

// ---------------------------------------------------------------------------
// Kernel 1: zero per-row counters, poison inverse index with -1
// ---------------------------------------------------------------------------
__global__ void moe_init_kernel(int* __restrict__ cnt, int* __restrict__ inv,
                                int N, int NK) {
    int i = blockIdx.x * blockDim.x + threadIdx.x;
    if (i < N)  cnt[i] = 0;
    if (i < NK) inv[i] = -1;
}

// ---------------------------------------------------------------------------
// Kernel 2: invert the scatter: for each slot i, record it under its row b.
// ---------------------------------------------------------------------------
__global__ void moe_build_inv_kernel(const int* __restrict__ bidx,
                                     int* __restrict__ cnt,
                                     int* __restrict__ inv,
                                     int NK, int K) {
    int i = blockIdx.x * blockDim.x + threadIdx.x;
    if (i >= NK) return;
    int b = bidx[i];
    int p = atomicAdd(&cnt[b], 1);
    if (p < K) inv[b * K + p] = i;
}

// ---------------------------------------------------------------------------
// Kernel 3 (K==2 fast path): 16-row x 128-col output tile per block.
// Stage 32 gathered source rows to LDS (coalesced NT b128), then 8 waves each
// build one 16x16 f32 tile with 8x V_WMMA_F32_16X16X4_F32 (exact f32 math).
// ---------------------------------------------------------------------------
#define TROWS 16
#define TSRC  32
#define TCOLS 128

__launch_bounds__(256)
__global__ void moe_combine_wmma_kernel(const float* __restrict__ S,
                                        const float* __restrict__ gates,
                                        const int* __restrict__ inv,
                                        float* __restrict__ out, int D) {
    __shared__ float lds[TSRC][TCOLS];   // 16 KB
    __shared__ float g_sh[TSRC];
    __shared__ int   idx_sh[TSRC];

    const int t  = threadIdx.x;          // 0..255
    const int r0 = blockIdx.y * TROWS;
    const int c0 = blockIdx.x * TCOLS;

    // Gather + deterministically order the 2 slots of each of the 16 rows.
    if (t < TROWS) {
        int i0 = inv[(size_t)(r0 + t) * 2 + 0];
        int i1 = inv[(size_t)(r0 + t) * 2 + 1];
        if (i1 >= 0 && (i0 < 0 || i1 < i0)) { int tmp = i0; i0 = i1; i1 = tmp; }
        idx_sh[2 * t + 0] = (i0 >= 0) ? i0 : 0;
        idx_sh[2 * t + 1] = (i1 >= 0) ? i1 : 0;
        g_sh[2 * t + 0]   = (i0 >= 0) ? gates[i0] : 0.0f;
        g_sh[2 * t + 1]   = (i1 >= 0) ? gates[i1] : 0.0f;
    }
    __syncthreads();

    const int  wave = t >> 5;
    const int  lane = t & 31;
    const int  cc   = wave * 16;       // 8 waves cover 128 cols
    const int  lcol = lane & 15;
    const bool hi   = lane >= 16;

    // Per-lane A-gate prefetch: each lane's A operand is nonzero for at most
    // ONE of the 8 K-chunks, and its two values are always g_sh[2*lcol+0/1].
    // (low half: lcol even, chunk g = lcol>>1; high half: lcol odd, same g.)
    // One ds_load_b64 here, then pure v_cndmask per chunk -> no predicated
    // LDS loads / EXEC churn around the WMMAs.
    const bool par_ok = ((lcol & 1) == (hi ? 1 : 0));
    const int  mygrp  = lcol >> 1;
    const float gA = g_sh[2 * lcol + 0];
    const float gB = g_sh[2 * lcol + 1];

    // Cooperative stage: 32 rows x 128 cols = 1024 float4, 4 per thread.
    // Streamed once globally -> non-temporal loads (protect 192MB L2).
#pragma unroll
    for (int j = 0; j < (TSRC * TCOLS / 4) / 256; ++j) {
        int lin = j * 256 + t;
        int row = lin / (TCOLS / 4);
        int c4  = lin % (TCOLS / 4);
        const v4f* p = (const v4f*)(S + (size_t)idx_sh[row] * D + c0);
        v4f v = __builtin_nontemporal_load(p + c4);
        *(v4f*)&lds[row][c4 * 4] = v;
    }
    __syncthreads();

    v8f C = {0.f, 0.f, 0.f, 0.f, 0.f, 0.f, 0.f, 0.f};

#if USE_WMMA
    // A is 16x32 block-diagonal (2x4 blocks of gates); split into 8 K=4 chunks
    // so each g*s product appears exactly once -> exact f32 result.
    // A layout (16x4 f32): lanes0-15 M=lane {K0,K1}; lanes16-31 M=lane-16 {K2,K3}.
    // B layout (4x16 f32): VGPR0 = {K0 | K2} rows, VGPR1 = {K1 | K3} rows.
#pragma unroll
    for (int g = 0; g < 8; ++g) {
        const bool sel  = par_ok && (mygrp == g);
        const int  base = 4 * g + (hi ? 2 : 0);  // B slot rows for this half
        v2f A;
        A.x = sel ? gA : 0.0f;
        A.y = sel ? gB : 0.0f;
        v2f B;
        B.x = lds[base + 0][cc + lcol];
        B.y = lds[base + 1][cc + lcol];
        C = __builtin_amdgcn_wmma_f32_16x16x4_f32(
                /*neg_a=*/false, A, /*neg_b=*/false, B,
                /*c_mod=*/(short)0, C, /*reuse_a=*/false, /*reuse_b=*/false);
    }
#else
    // VALU fallback with identical C layout (VGPR v -> rows v / v+8).
#pragma unroll
    for (int v = 0; v < 8; ++v) {
        int m = v + (hi ? 8 : 0);
        C[v] = g_sh[2 * m + 0] * lds[2 * m + 0][cc + lcol]
             + g_sh[2 * m + 1] * lds[2 * m + 1][cc + lcol];
    }
#endif

    // C/D layout: VGPR v -> (M = v or v+8, N = lane%16). 64B segments per half-
    // wave, written once and never reread -> non-temporal stores.
#pragma unroll
    for (int v = 0; v < 8; ++v) {
        int m = v + (hi ? 8 : 0);
        __builtin_nontemporal_store(
            C[v], &out[(size_t)(r0 + m) * D + (size_t)(c0 + cc + lcol)]);
    }
}

// ---------------------------------------------------------------------------
// Generic fallback (any K, any D%4==0): one block per output row, float4 axpy.
// ---------------------------------------------------------------------------
__global__ void moe_combine_scalar_kernel(const float* __restrict__ S,
                                          const float* __restrict__ gates,
                                          const int* __restrict__ inv,
                                          float* __restrict__ out,
                                          int D, int K) {
    const int b = blockIdx.x;
    // Deterministic slot order: insertion sort indices ascending (-1 last).
    int   idx[16];
    float gv[16];
    int kk = (K > 16) ? 16 : K;
    for (int k = 0; k < kk; ++k) idx[k] = inv[(size_t)b * K + k];
    for (int a = 1; a < kk; ++a) {
        int v = idx[a], j = a - 1;
        while (j >= 0 && ((unsigned)idx[j] > (unsigned)v)) { idx[j + 1] = idx[j]; --j; }
        idx[j + 1] = v;
    }
    for (int k = 0; k < kk; ++k) gv[k] = (idx[k] >= 0) ? gates[idx[k]] : 0.0f;

    const int D4 = D >> 2;
    for (int c = threadIdx.x; c < D4; c += blockDim.x) {
        v4f acc = {0.f, 0.f, 0.f, 0.f};
        for (int k = 0; k < kk; ++k) {
            if (idx[k] >= 0) {
                const v4f* p = (const v4f*)(S + (size_t)idx[k] * D);
                v4f x = __builtin_nontemporal_load(p + c);
                acc.x += gv[k] * x.x; acc.y += gv[k] * x.y;
                acc.z += gv[k] * x.z; acc.w += gv[k] * x.w;
            }
        }
        __builtin_nontemporal_store(acc, (v4f*)out + (size_t)b * D4 + c);
    }
}

// ---------------------------------------------------------------------------
// Host launcher
// ---------------------------------------------------------------------------
extern "C" void kernel_launch(void* const* d_in, const int* in_sizes, int n_in,
                              void* d_out, int out_size, void* d_ws, size_t ws_size,
                              hipStream_t stream) {
    const float* S     = (const float*)d_in[0];   // [N*K, D]
    const float* gates = (const float*)d_in[1];   // [N*K]
    const int*   bidx  = (const int*)d_in[2];     // [N*K]
    float*       out   = (float*)d_out;           // [N, D]

    const int NK = in_sizes[2];
    const int D  = in_sizes[0] / NK;              // 4096
    const int N  = out_size / D;                  // 8192
    const int K  = NK / N;                        // 2

    int* cnt = (int*)d_ws;                        // N ints
    int* inv = cnt + N;                           // N*K ints

    const int initN = (N > NK) ? N : NK;
    moe_init_kernel<<<(initN + 255) / 256, 256, 0, stream>>>(cnt, inv, N, NK);
    moe_build_inv_kernel<<<(NK + 255) / 256, 256, 0, stream>>>(bidx, cnt, inv, NK, K);

    if (K == 2 && (D % TCOLS) == 0 && (N % TROWS) == 0) {
        dim3 grid(D / TCOLS, N / TROWS);          // (32, 512)
        moe_combine_wmma_kernel<<<grid, 256, 0, stream>>>(S, gates, inv, out, D);
    } else {
        moe_combine_scalar_kernel<<<N, 256, 0, stream>>>(S, gates, inv, out, D, K);
    }
}